// S6_26611617366353
// MI455X (gfx1250) — compile-verified
//
#include <hip/hip_runtime.h>
#include <hip/hip_bf16.h>
#include <stdint.h>

// Problem dims (fixed by the reference)
#define BSZ   16
#define LSEQ  2048
#define DDIM  128
#define NST   16
#define MROWS (BSZ * LSEQ)          // 32768 flattened (b,l) rows
#define NTILES_N 10                 // 160 fused cols: [W1(8 tiles) | W2 | W3]

typedef float v2f __attribute__((ext_vector_type(2)));
typedef float v8f __attribute__((ext_vector_type(8)));

// ---- CDNA5 async global->LDS copy (ASYNCcnt path, CDNA5 ISA) ----
// The LDS destination is passed as a real pointer and cast to u32: per the ISA
// flat-aperture rule (10.2) the low 32 bits of a
// generic LDS pointer are the LDS byte address. Crucially, the ptrtoint makes
// the shared array ESCAPE into the asm, so LLVM cannot fold subsequent LDS
// loads to undef (which it did in the previous round, where only a plain
// integer offset was passed and the ds reads were eliminated).
__device__ __forceinline__ void async_b128_to_lds(void* lds_ptr,
                                                  const void* gaddr) {
    uint32_t l = (uint32_t)(uintptr_t)lds_ptr;   // LDS byte address (addr[31:0])
    uint64_t g = (uint64_t)(uintptr_t)gaddr;
    asm volatile("global_load_async_to_lds_b128 %0, %1, off"
                 :: "v"(l), "v"(g) : "memory");
}
__device__ __forceinline__ void wait_async0() {
    asm volatile("s_wait_asynccnt 0x0" ::: "memory");
}

// -----------------------------------------------------------------------------
// Kernel 1: fused projection GEMM  P = x @ [W1|W2|W3] + [b1|b2|b3]
//   grid = (256 M-blocks, 10 N-tiles); block = 256 threads = 8 waves.
//   All 8 waves share one 16-col weight tile, async-staged into LDS (8 KB).
//   Wave w computes rows [ (bx*8+w)*16, +16 ) via 32x V_WMMA_F32_16X16X4_F32.
//   Epilogue (scalar branch on blockIdx.y):
//     N-tiles 0..7 -> delta = softplus(P);  tile 8 -> Bm;  tile 9 -> Cm
// -----------------------------------------------------------------------------
__global__ __launch_bounds__(256) void s6_proj_gemm(
    const float* __restrict__ x,
    const float* __restrict__ W1, const float* __restrict__ b1,
    const float* __restrict__ W2, const float* __restrict__ b2,
    const float* __restrict__ W3, const float* __restrict__ b3,
    float* __restrict__ delta_ws, float* __restrict__ Bm_ws,
    float* __restrict__ Cm_ws)
{
    __shared__ float lW[DDIM * NST];   // weight tile, row-major [K=128][N=16]

    const int t      = threadIdx.x;
    const int lane   = t & 31;
    const int warp   = t >> 5;
    const int tile_n = blockIdx.y;     // uniform per block (scalar branches)
    const int n0     = tile_n * 16;

    const float* Wp; const float* bp; int ld4; int c4;
    if (tile_n < 8)       { Wp = W1; bp = b1; ld4 = DDIM / 4; c4 = n0 >> 2; }
    else if (tile_n == 8) { Wp = W2; bp = b2; ld4 = NST / 4;  c4 = 0; }
    else                  { Wp = W3; bp = b3; ld4 = NST / 4;  c4 = 0; }

    // Async-stage the 128x16 weight tile: 512 float4, 2 per thread.
    {
        const float4* Wsrc = (const float4*)Wp;
        #pragma unroll
        for (int i = t; i < DDIM * NST / 4; i += 256) {
            const int k = i >> 2, seg = i & 3;
            async_b128_to_lds(lW + (size_t)i * 4,
                              Wsrc + (size_t)k * ld4 + c4 + seg);
        }
    }
    wait_async0();
    __syncthreads();

    const int tile_m = blockIdx.x * 8 + warp;
    const int nl     = lane & 15;          // C/D col, and A row (M) for this lane
    const int m_half = lane >> 4;          // 0: M 0..7 / K 0,1 ; 1: M 8..15 / K 2,3
    const int koff   = m_half * 2;

    const float* arow = x + (size_t)(tile_m * 16 + nl) * DDIM;

    union { v8f v; float f[8]; } acc;
    acc.v = (v8f){0.f, 0.f, 0.f, 0.f, 0.f, 0.f, 0.f, 0.f};

    #pragma unroll 8
    for (int k0 = 0; k0 < DDIM; k0 += 4) {
        // A 16x4 f32: lanes 0-15 hold K=k0,k0+1; lanes 16-31 hold K=k0+2,k0+3
        const float2 av = *(const float2*)(arow + k0 + koff);   // global_load_b64
        v2f a; a.x = av.x; a.y = av.y;
        // B 4x16 f32 from LDS (mirrored K split across lane halves)
        v2f b;
        b.x = lW[(k0 + koff)     * NST + nl];                   // ds_load_b32
        b.y = lW[(k0 + koff + 1) * NST + nl];
        // (neg_a, A, neg_b, B, c_mod, C, reuse_a, reuse_b)
        acc.v = __builtin_amdgcn_wmma_f32_16x16x4_f32(
            false, a, false, b, (short)0, acc.v, false, false);
    }

    const float bias = bp[nl];

    // C/D 16x16 f32 layout: vgpr r -> M = r + 8*m_half, lane&15 -> N
    if (tile_n < 8) {
        #pragma unroll
        for (int r = 0; r < 8; ++r) {
            const int m = tile_m * 16 + m_half * 8 + r;
            const float v = acc.f[r] + bias;
            const float sp = (v > 20.f) ? v : __logf(1.f + __expf(v));
            delta_ws[(size_t)m * DDIM + n0 + nl] = sp;
        }
    } else if (tile_n == 8) {
        #pragma unroll
        for (int r = 0; r < 8; ++r) {
            const int m = tile_m * 16 + m_half * 8 + r;
            Bm_ws[(size_t)m * NST + nl] = acc.f[r] + bias;
        }
    } else {
        #pragma unroll
        for (int r = 0; r < 8; ++r) {
            const int m = tile_m * 16 + m_half * 8 + r;
            Cm_ws[(size_t)m * NST + nl] = acc.f[r] + bias;
        }
    }
}

// -----------------------------------------------------------------------------
// Kernel 2: streaming state update + contraction over N (HBM-bound on h)
//   y[bl,d] = sum_n Cm[bl,n] * ( exp(delta[bl,d]*A[d,n]) * h[bl,d,n]
//                                + x[bl,d]*delta[bl,d]*Bm[bl,n] )
//   256 threads = 2 (b,l) pairs x 128 d. A async-staged to LDS (8 KB);
//   h read as coalesced 64 B/thread float4s (the bandwidth-critical stream).
// -----------------------------------------------------------------------------
__global__ __launch_bounds__(256) void s6_state_update(
    const float* __restrict__ x,
    const float* __restrict__ A,
    const float* __restrict__ h,
    const float* __restrict__ delta_ws,
    const float* __restrict__ Bm_ws,
    const float* __restrict__ Cm_ws,
    float* __restrict__ y)
{
    // single LDS allocation: A[2048] | Bm[2][16] | Cm[2][16]
    __shared__ float smem[DDIM * NST + 4 * NST];
    float* lA = smem;
    float* lB = smem + DDIM * NST;
    float* lC = lB + 2 * NST;

    const int  t   = threadIdx.x;
    const long bl0 = (long)blockIdx.x * 2;

    // Async-stage A (512 float4), 2 per thread.
    #pragma unroll
    for (int i = t; i < DDIM * NST / 4; i += 256) {
        async_b128_to_lds(lA + (size_t)i * 4, (const float4*)A + i);
    }
    // Stage Bm/Cm for the two (b,l) pairs: 8 float4 each (compiler-visible).
    if (t < 8) {
        ((float4*)lB)[t] = ((const float4*)Bm_ws)[bl0 * 4 + t];
    } else if (t < 16) {
        const int q = t - 8;
        ((float4*)lC)[q] = ((const float4*)Cm_ws)[bl0 * 4 + q];
    }
    wait_async0();
    __syncthreads();

    const int  p  = t >> 7;            // which (b,l) pair
    const int  d  = t & 127;
    const long bd = (bl0 + p) * DDIM + d;

    const float dl = delta_ws[bd];
    const float xd = x[bd] * dl;

    const float4* __restrict__ h4 = (const float4*)(h + bd * NST);
    const float4* a4 = (const float4*)(lA + d * NST);
    const float4* B4 = (const float4*)(lB + p * NST);
    const float4* C4 = (const float4*)(lC + p * NST);

    __builtin_prefetch(h4, 0, 1);      // gfx1250 global_prefetch hint

    float acc = 0.f;
    #pragma unroll
    for (int j = 0; j < NST / 4; ++j) {
        const float4 hv = h4[j];
        const float4 av = a4[j];
        const float4 bv = B4[j];
        const float4 cv = C4[j];
        acc += cv.x * (__expf(dl * av.x) * hv.x + xd * bv.x);
        acc += cv.y * (__expf(dl * av.y) * hv.y + xd * bv.y);
        acc += cv.z * (__expf(dl * av.z) * hv.z + xd * bv.z);
        acc += cv.w * (__expf(dl * av.w) * hv.w + xd * bv.w);
    }

    y[bd] = acc;
}

// -----------------------------------------------------------------------------
// Host-side launch
// -----------------------------------------------------------------------------
extern "C" void kernel_launch(void* const* d_in, const int* in_sizes, int n_in,
                              void* d_out, int out_size, void* d_ws, size_t ws_size,
                              hipStream_t stream) {
    (void)in_sizes; (void)n_in; (void)out_size; (void)ws_size;

    const float* x  = (const float*)d_in[0];
    const float* W1 = (const float*)d_in[1];
    const float* b1 = (const float*)d_in[2];
    const float* W2 = (const float*)d_in[3];
    const float* b2 = (const float*)d_in[4];
    const float* W3 = (const float*)d_in[5];
    const float* b3 = (const float*)d_in[6];
    const float* A  = (const float*)d_in[7];
    const float* h  = (const float*)d_in[8];
    float* y = (float*)d_out;

    // Workspace (floats): delta [32768*128] | Bm [32768*16] | Cm [32768*16]
    float* ws       = (float*)d_ws;
    float* delta_ws = ws;
    float* Bm_ws    = delta_ws + (size_t)MROWS * DDIM;
    float* Cm_ws    = Bm_ws + (size_t)MROWS * NST;

    // Kernel 1: 256 M-blocks (8 waves x 16 rows) x 10 N-tiles
    dim3 g1(MROWS / (16 * 8), NTILES_N, 1);
    s6_proj_gemm<<<g1, 256, 0, stream>>>(
        x, W1, b1, W2, b2, W3, b3, delta_ws, Bm_ws, Cm_ws);

    // Kernel 2: 32768 (b,l) pairs, 2 per block
    s6_state_update<<<MROWS / 2, 256, 0, stream>>>(
        x, A, h, delta_ws, Bm_ws, Cm_ws, y);
}